// DAGMixer_8306466750965
// MI455X (gfx1250) — compile-verified
//
#include <hip/hip_runtime.h>
#include <hip/hip_bf16.h>
#include <stdint.h>

// ---------------------------------------------------------------------------
// DAGMixer: fused gate-GEMV(N=1) + sigmoid + lerp.  Pure HBM-bandwidth kernel
// (~402 MB moved, ~17us floor at 23.3 TB/s).  One block per (b,t) row.
// CDNA5 specifics used:
//   * Tensor Data Mover (tensor_load_to_lds) to stage W_gate (16KB) into LDS,
//     waited with s_wait_tensorcnt, overlapping the streaming global loads.
//   * wave32 shuffle reduction (warpSize==32 on gfx1250).
//   * nontemporal b128 loads/stores for the 3 streaming tensors so the 192MB
//     L2 stays dedicated to the heavily-reused W_gate.
// WMMA deliberately unused: GEMV with N=1 (0.6 FLOP/byte) — 16x16x4 WMMA
// would waste 16x the multiplies and force lane-strided (uncoalesced) A
// loads on a kernel limited only by HBM bandwidth.
// ---------------------------------------------------------------------------

typedef float        v4f   __attribute__((ext_vector_type(4)));
typedef unsigned int u32x4 __attribute__((ext_vector_type(4)));
typedef int          i32x4 __attribute__((ext_vector_type(4)));
typedef int          i32x8 __attribute__((ext_vector_type(8)));

#define HDIM    2048
#define ROWS    16384          // B*T = 8*2048
#define THREADS 256            // 8 wave32s
#define V4_PER_ROW (HDIM / 4)  // 512

#if defined(__has_builtin)
#if __has_builtin(__builtin_amdgcn_tensor_load_to_lds) && \
    __has_builtin(__builtin_amdgcn_s_wait_tensorcnt)
#define DAGMIX_USE_TDM 1
#endif
#endif

__global__ __launch_bounds__(THREADS) void DAGMixer_kernel(
    const float* __restrict__ orig,    // [ROWS, HDIM]
    const float* __restrict__ dag,     // [ROWS, HDIM]
    const float* __restrict__ wgate,   // [2*HDIM]  (W_o then W_d)
    const float* __restrict__ bgate,   // [1]
    float* __restrict__ out)           // mixed [ROWS*HDIM] then gates [ROWS]
{
    __shared__ float w_lds[2 * HDIM];          // 16 KB: W_o | W_d
    __shared__ v4f   red4[2];                  // 8 per-wave partials, 16B-aligned

    const int    tid  = threadIdx.x;
    const int    row  = blockIdx.x;
    const size_t base = (size_t)row * HDIM;

    const v4f* o4 = (const v4f*)(orig + base);
    const v4f* d4 = (const v4f*)(dag + base);

    // Issue the streaming loads first (nontemporal: 400MB >> L2) so they are
    // in flight while the TDM stages W_gate into LDS.
    v4f o0  = __builtin_nontemporal_load(o4 + tid);
    v4f o1  = __builtin_nontemporal_load(o4 + tid + THREADS);
    v4f g0v = __builtin_nontemporal_load(d4 + tid);
    v4f g1v = __builtin_nontemporal_load(d4 + tid + THREADS);

#if defined(DAGMIX_USE_TDM)
    if (tid == 0) {
        // Build Tensor DMA descriptor (D#): 1-D tile of 4096 f32 -> LDS.
        const uint64_t ga     = (uint64_t)(uintptr_t)wgate;
        const uint32_t ldsOff = (uint32_t)(uintptr_t)(void*)w_lds; // LDS byte offset

        // Group 0: count=1 (valid), lds_addr, global_addr[56:0], type=2
        u32x4 dg0;
        dg0.x = 1u;                                            // count=1, user mode
        dg0.y = ldsOff;                                        // lds_addr (bytes)
        dg0.z = (uint32_t)ga;                                  // global_addr[31:0]
        dg0.w = ((uint32_t)(ga >> 32) & 0x01FFFFFFu)           // global_addr[56:32]
                | 0x80000000u;                                 // type=2 ("image")

        // Group 1: data_size=4B, tensor_dim0=4096, tensor_dim1=1,
        //          tile_dim0=4096, tile_dim1=1, stride0=4096. No multicast.
        i32x8 dg1;
        dg1[0] = (int)(2u << 16);       // workgroup_mask=0, data_size=2 (4 bytes)
        dg1[1] = (int)(4096u << 16);    // tensor_dim0[15:0] in bits 63:48
        dg1[2] = (int)(1u << 16);       // tensor_dim0[31:16]=0 | tensor_dim1=1
        dg1[3] = (int)(4096u << 16);    // tensor_dim1[31:16]=0 | tile_dim0=4096
        dg1[4] = 1;                     // tile_dim1=1, tile_dim2=0
        dg1[5] = 4096;                  // tensor_dim0_stride[31:0]
        dg1[6] = 0;                     // stride0[47:32]=0, tensor_dim1_stride lo=0
        dg1[7] = 0;

        i32x4 dgz = {0, 0, 0, 0};       // groups 2/3: tile_dim3/4 = 0 (unused)
#if defined(__clang_major__) && (__clang_major__ >= 23)
        i32x8 dgz8 = {0, 0, 0, 0, 0, 0, 0, 0};
        __builtin_amdgcn_tensor_load_to_lds(dg0, dg1, dgz, dgz, dgz8, 0);
#else
        __builtin_amdgcn_tensor_load_to_lds(dg0, dg1, dgz, dgz, 0);
#endif
        __builtin_amdgcn_s_wait_tensorcnt(0);  // TDM done before barrier signal
    }
#else
    for (int i = tid; i < 2 * HDIM; i += THREADS) w_lds[i] = wgate[i];
#endif
    __syncthreads();

    // Per-thread partial of  <orig,W_o> + <dag,W_d>  (LDS reads conflict-free:
    // 32 lanes x 16B consecutive).
    const v4f* w4  = (const v4f*)w_lds;
    const v4f wo0 = w4[tid];
    const v4f wo1 = w4[tid + THREADS];
    const v4f wd0 = w4[V4_PER_ROW + tid];
    const v4f wd1 = w4[V4_PER_ROW + tid + THREADS];

    float p = o0.x * wo0.x + o0.y * wo0.y + o0.z * wo0.z + o0.w * wo0.w
            + o1.x * wo1.x + o1.y * wo1.y + o1.z * wo1.z + o1.w * wo1.w
            + g0v.x * wd0.x + g0v.y * wd0.y + g0v.z * wd0.z + g0v.w * wd0.w
            + g1v.x * wd1.x + g1v.y * wd1.y + g1v.z * wd1.z + g1v.w * wd1.w;

    // wave32 butterfly reduction -> one partial per wave
    #pragma unroll
    for (int off = 16; off > 0; off >>= 1) p += __shfl_xor(p, off, 32);
    if ((tid & 31) == 0) ((float*)red4)[tid >> 5] = p;
    __syncthreads();

    // Every thread finishes the 8-way sum itself (two aligned ds_load_b128)
    // and computes its own sigmoid: kills the broadcast barrier and the
    // thread-0 serialization; 256 v_exp_f32/block is noise on the TRANS pipe.
    const v4f r0 = red4[0];
    const v4f r1 = red4[1];
    const float s = ((r0.x + r0.y) + (r0.z + r0.w))
                  + ((r1.x + r1.y) + (r1.z + r1.w)) + bgate[0];
    const float g = 1.0f / (1.0f + __expf(-s));

    if (tid == 0) out[(size_t)ROWS * HDIM + row] = g;   // gate_values output

    v4f m0 = o0 + g * (g0v - o0);
    v4f m1 = o1 + g * (g1v - o1);

    v4f* out4 = (v4f*)(out + base);
    __builtin_nontemporal_store(m0, out4 + tid);
    __builtin_nontemporal_store(m1, out4 + tid + THREADS);
}

extern "C" void kernel_launch(void* const* d_in, const int* in_sizes, int n_in,
                              void* d_out, int out_size, void* d_ws, size_t ws_size,
                              hipStream_t stream) {
    (void)in_sizes; (void)n_in; (void)out_size; (void)d_ws; (void)ws_size;
    const float* orig = (const float*)d_in[0];
    const float* dag  = (const float*)d_in[1];
    const float* wg   = (const float*)d_in[2];
    const float* bg   = (const float*)d_in[3];
    float*       out  = (float*)d_out;
    DAGMixer_kernel<<<dim3(ROWS), dim3(THREADS), 0, stream>>>(orig, dag, wg, bg, out);
}